// StandGATX_15839839387782
// MI455X (gfx1250) — compile-verified
//
#include <hip/hip_runtime.h>
#include <hip/hip_bf16.h>

#define NNODES 50000
#define NEDGES 800000
#define GFEAT  256
#define GHID   64
#define GCLASS 40
#define GHEADS 4
#define GHDIM  16
#define NEG_SLOPE 0.2f

typedef float v2f __attribute__((ext_vector_type(2)));
typedef float v8f __attribute__((ext_vector_type(8)));

// ---------------------------------------------------------------------------
// float atomic max via IEEE int ordering (handles negatives)
// ---------------------------------------------------------------------------
__device__ __forceinline__ void atomic_max_f32(float* addr, float val) {
  if (val >= 0.0f) {
    atomicMax((int*)addr, __float_as_int(val));
  } else {
    atomicMin((unsigned int*)addr, __float_as_uint(val));
  }
}

// ---------------------------------------------------------------------------
// C[M,NC] = A[M,K] * W[K,NC], full-precision f32 WMMA (16x16 tiles, K-step 4).
// One wave32 owns a full 16-row strip of C (all NT column tiles), so the A
// fragment is loaded ONCE per k-step and reused across NT WMMAs.
// M must be a multiple of 16 (50000 = 3125*16).
// A layout per ISA: lanes 0-15 rows M=0..15, v0=K(+0)/K(+2), v1=K(+1)/K(+3).
// B layout: v0 = rows k+0 (lanes 0-15) / k+2 (lanes 16-31), v1 = k+1 / k+3.
// ---------------------------------------------------------------------------
template<int K, int NC>
__global__ __launch_bounds__(256) void gemm_wmma_f32(
    const float* __restrict__ A, const float* __restrict__ W,
    float* __restrict__ C, int M) {
  constexpr int NT = (NC + 15) / 16;
  const int wave = (int)((blockIdx.x * blockDim.x + threadIdx.x) >> 5);
  const int lane = (int)(threadIdx.x & 31);
  const int mtiles = M >> 4;
  if (wave >= mtiles) return;        // wave-uniform exit: EXEC stays full for WMMA
  const int mt  = wave;
  const int l16 = lane & 15;
  const int hi  = lane >> 4;         // 0: lanes 0-15, 1: lanes 16-31
  const int row = (mt << 4) + l16;
  const float* __restrict__ arow = A + (size_t)row * K;

  // per-N-tile column / mask (handles NC=40 partial tile without divergence)
  int   ccol[NT];
  float cmask[NT];
#pragma unroll
  for (int nt = 0; nt < NT; ++nt) {
    const int gcol = (nt << 4) + l16;
    ccol[nt]  = (gcol < NC) ? gcol : 0;
    cmask[nt] = (gcol < NC) ? 1.0f : 0.0f;
  }

  v8f acc[NT];
#pragma unroll
  for (int nt = 0; nt < NT; ++nt) acc[nt] = {};

  for (int k = 0; k < K; k += 4) {
    v2f a;
    a.x = arow[k + 2 * hi + 0];
    a.y = arow[k + 2 * hi + 1];
    const float* __restrict__ w0 = W + (size_t)(k + 2 * hi + 0) * NC;
    const float* __restrict__ w1 = W + (size_t)(k + 2 * hi + 1) * NC;
#pragma unroll
    for (int nt = 0; nt < NT; ++nt) {
      v2f b;
      b.x = w0[ccol[nt]] * cmask[nt];
      b.y = w1[ccol[nt]] * cmask[nt];
      acc[nt] = __builtin_amdgcn_wmma_f32_16x16x4_f32(
          /*neg_a=*/false, a, /*neg_b=*/false, b,
          /*c_mod=*/(short)0, acc[nt], /*reuse_a=*/false, /*reuse_b=*/false);
    }
  }

  // C/D layout: VGPR r -> row mt*16 + r (lanes 0-15) / mt*16 + 8 + r (lanes 16-31)
  const int rbase = (mt << 4) + (hi << 3);
#pragma unroll
  for (int nt = 0; nt < NT; ++nt) {
    const int gcol = (nt << 4) + l16;
    if (gcol < NC) {
#pragma unroll
      for (int r = 0; r < 8; ++r)
        C[(size_t)(rbase + r) * NC + gcol] = acc[nt][r];
    }
  }
}

// ---------------------------------------------------------------------------
// alpha_s[n,h] = sum_c h[n,h,c]*a_src[h,c]   (and same for a_dst)
// ---------------------------------------------------------------------------
template<int H, int C>
__global__ __launch_bounds__(256) void alpha_kernel(
    const float* __restrict__ hfeat, const float* __restrict__ a_src,
    const float* __restrict__ a_dst, float* __restrict__ as,
    float* __restrict__ ad, int n) {
  int t = blockIdx.x * blockDim.x + threadIdx.x;
  if (t >= n * H) return;
  const int node = t / H, hd = t % H;
  const float* hp = hfeat + (size_t)node * (H * C) + hd * C;
  float s = 0.f, d = 0.f;
#pragma unroll
  for (int c = 0; c < C; ++c) {
    s += hp[c] * a_src[hd * C + c];
    d += hp[c] * a_dst[hd * C + c];
  }
  as[t] = s;
  ad[t] = d;
}

__global__ __launch_bounds__(256) void fill_kernel(float* p, float v, int n) {
  int t = blockIdx.x * blockDim.x + threadIdx.x;
  if (t < n) p[t] = v;
}

// ---------------------------------------------------------------------------
// Pass 1: edge logits e = leaky_relu(as[src]+ad[dst]); segment max via atomics
// Edge t < E comes from edge_index; t >= E is self-loop (src=dst=t-E).
// ---------------------------------------------------------------------------
template<int H>
__global__ __launch_bounds__(256) void edge_logits_kernel(
    const long long* __restrict__ ei, int E, int ET,
    const float* __restrict__ as, const float* __restrict__ ad,
    float* __restrict__ ebuf, float* __restrict__ emax) {
  int t = blockIdx.x * blockDim.x + threadIdx.x;
  if (t >= ET) return;
  int s, d;
  if (t < E) { s = (int)ei[t]; d = (int)ei[(size_t)E + t]; }
  else       { s = d = t - E; }
#pragma unroll
  for (int h = 0; h < H; ++h) {
    float e = as[s * H + h] + ad[d * H + h];
    e = (e > 0.f) ? e : NEG_SLOPE * e;        // leaky relu
    ebuf[(size_t)t * H + h] = e;
    atomic_max_f32(&emax[d * H + h], e);
  }
}

// Pass 2: e_exp = exp(e - max[dst]); segment sum via atomics
template<int H>
__global__ __launch_bounds__(256) void edge_exp_kernel(
    const long long* __restrict__ ei, int E, int ET,
    float* __restrict__ ebuf, const float* __restrict__ emax,
    float* __restrict__ denom) {
  int t = blockIdx.x * blockDim.x + threadIdx.x;
  if (t >= ET) return;
  int d = (t < E) ? (int)ei[(size_t)E + t] : t - E;
#pragma unroll
  for (int h = 0; h < H; ++h) {
    float ex = expf(ebuf[(size_t)t * H + h] - emax[d * H + h]);
    ebuf[(size_t)t * H + h] = ex;
    atomicAdd(&denom[d * H + h], ex);
  }
}

// Pass 3: agg[dst] += h[src] * (e_exp / denom[dst]); one thread per (edge, 4 chans)
template<int H, int C>
__global__ __launch_bounds__(256) void edge_msg_kernel(
    const long long* __restrict__ ei, int E, int ET,
    const float* __restrict__ hfeat, const float* __restrict__ ebuf,
    const float* __restrict__ denom, float* __restrict__ agg) {
  constexpr int OC = H * C;
  constexpr int G  = OC / 4;
  int tid = blockIdx.x * blockDim.x + threadIdx.x;
  if (tid >= ET * G) return;
  const int t = tid / G;
  const int g = tid % G;
  int s, d;
  if (t < E) { s = (int)ei[t]; d = (int)ei[(size_t)E + t]; }
  else       { s = d = t - E; }
  const int j = g * 4;
  const int h = j / C;
  const float alpha = ebuf[(size_t)t * H + h] / (denom[d * H + h] + 1e-16f);
  const float4 hv = *reinterpret_cast<const float4*>(hfeat + (size_t)s * OC + j);
  float* o = agg + (size_t)d * OC + j;
  atomicAdd(o + 0, hv.x * alpha);
  atomicAdd(o + 1, hv.y * alpha);
  atomicAdd(o + 2, hv.z * alpha);
  atomicAdd(o + 3, hv.w * alpha);
}

template<bool RELU>
__global__ __launch_bounds__(256) void finalize_kernel(
    const float* __restrict__ agg, const float* __restrict__ bias,
    float* __restrict__ out, int n, int OC) {
  int t = blockIdx.x * blockDim.x + threadIdx.x;
  if (t >= n * OC) return;
  float v = agg[t] + bias[t % OC];
  out[t] = RELU ? fmaxf(v, 0.f) : v;
}

// ---------------------------------------------------------------------------
static inline int cdiv(long long a, int b) { return (int)((a + b - 1) / b); }

extern "C" void kernel_launch(void* const* d_in, const int* in_sizes, int n_in,
                              void* d_out, int out_size, void* d_ws, size_t ws_size,
                              hipStream_t stream) {
  const float*     x   = (const float*)d_in[0];
  const long long* ei  = (const long long*)d_in[1];
  const float*     W1  = (const float*)d_in[2];
  const float*     aS1 = (const float*)d_in[3];
  const float*     aD1 = (const float*)d_in[4];
  const float*     b1  = (const float*)d_in[5];
  const float*     W2  = (const float*)d_in[6];
  const float*     aS2 = (const float*)d_in[7];
  const float*     aD2 = (const float*)d_in[8];
  const float*     b2  = (const float*)d_in[9];
  const float*     W3  = (const float*)d_in[10];
  const float*     aS3 = (const float*)d_in[11];
  const float*     aD3 = (const float*)d_in[12];
  const float*     b3  = (const float*)d_in[13];
  float* out = (float*)d_out;

  // workspace carve-up (floats)
  float* ws    = (float*)d_ws;
  float* hA    = ws;  ws += (size_t)NNODES * 64;           // transformed features
  float* featB = ws;  ws += (size_t)NNODES * 64;           // layer output / next input
  float* agg   = ws;  ws += (size_t)NNODES * 64;           // scatter accumulator
  float* as    = ws;  ws += (size_t)NNODES * GHEADS;
  float* ad    = ws;  ws += (size_t)NNODES * GHEADS;
  float* emax  = ws;  ws += (size_t)NNODES * GHEADS;
  float* den   = ws;  ws += (size_t)NNODES * GHEADS;
  float* ebuf  = ws;                                       // (E+N)*GHEADS

  const int ET = NEDGES + NNODES;
  const int TB = 256;
  const int g_edges = cdiv(ET, TB);
  const int g_gemm  = cdiv((long long)(NNODES / 16) * 32, TB);  // one wave per 16-row strip

  // ---------------- Layer 1: 256 -> 4x16, ReLU ----------------
  {
    gemm_wmma_f32<GFEAT, GHID><<<g_gemm, TB, 0, stream>>>(x, W1, hA, NNODES);
    alpha_kernel<GHEADS, GHDIM><<<cdiv(NNODES * GHEADS, TB), TB, 0, stream>>>(hA, aS1, aD1, as, ad, NNODES);
    fill_kernel<<<cdiv((long long)NNODES * 64, TB), TB, 0, stream>>>(agg, 0.0f, NNODES * 64);
    fill_kernel<<<cdiv(NNODES * GHEADS, TB), TB, 0, stream>>>(emax, -3.0e38f, NNODES * GHEADS);
    fill_kernel<<<cdiv(NNODES * GHEADS, TB), TB, 0, stream>>>(den, 0.0f, NNODES * GHEADS);
    edge_logits_kernel<GHEADS><<<g_edges, TB, 0, stream>>>(ei, NEDGES, ET, as, ad, ebuf, emax);
    edge_exp_kernel<GHEADS><<<g_edges, TB, 0, stream>>>(ei, NEDGES, ET, ebuf, emax, den);
    edge_msg_kernel<GHEADS, GHDIM><<<cdiv((long long)ET * 16, TB), TB, 0, stream>>>(ei, NEDGES, ET, hA, ebuf, den, agg);
    finalize_kernel<true><<<cdiv((long long)NNODES * 64, TB), TB, 0, stream>>>(agg, b1, featB, NNODES, 64);
  }

  // ---------------- Layer 2: 64 -> 4x16, ReLU ----------------
  {
    gemm_wmma_f32<GHID, GHID><<<g_gemm, TB, 0, stream>>>(featB, W2, hA, NNODES);
    alpha_kernel<GHEADS, GHDIM><<<cdiv(NNODES * GHEADS, TB), TB, 0, stream>>>(hA, aS2, aD2, as, ad, NNODES);
    fill_kernel<<<cdiv((long long)NNODES * 64, TB), TB, 0, stream>>>(agg, 0.0f, NNODES * 64);
    fill_kernel<<<cdiv(NNODES * GHEADS, TB), TB, 0, stream>>>(emax, -3.0e38f, NNODES * GHEADS);
    fill_kernel<<<cdiv(NNODES * GHEADS, TB), TB, 0, stream>>>(den, 0.0f, NNODES * GHEADS);
    edge_logits_kernel<GHEADS><<<g_edges, TB, 0, stream>>>(ei, NEDGES, ET, as, ad, ebuf, emax);
    edge_exp_kernel<GHEADS><<<g_edges, TB, 0, stream>>>(ei, NEDGES, ET, ebuf, emax, den);
    edge_msg_kernel<GHEADS, GHDIM><<<cdiv((long long)ET * 16, TB), TB, 0, stream>>>(ei, NEDGES, ET, hA, ebuf, den, agg);
    finalize_kernel<true><<<cdiv((long long)NNODES * 64, TB), TB, 0, stream>>>(agg, b2, featB, NNODES, 64);
  }

  // ---------------- Layer 3: 64 -> 40, heads=1, no ReLU ----------------
  {
    gemm_wmma_f32<GHID, GCLASS><<<g_gemm, TB, 0, stream>>>(featB, W3, hA, NNODES);
    alpha_kernel<1, GCLASS><<<cdiv(NNODES, TB), TB, 0, stream>>>(hA, aS3, aD3, as, ad, NNODES);
    fill_kernel<<<cdiv((long long)NNODES * GCLASS, TB), TB, 0, stream>>>(agg, 0.0f, NNODES * GCLASS);
    fill_kernel<<<cdiv(NNODES, TB), TB, 0, stream>>>(emax, -3.0e38f, NNODES);
    fill_kernel<<<cdiv(NNODES, TB), TB, 0, stream>>>(den, 0.0f, NNODES);
    edge_logits_kernel<1><<<g_edges, TB, 0, stream>>>(ei, NEDGES, ET, as, ad, ebuf, emax);
    edge_exp_kernel<1><<<g_edges, TB, 0, stream>>>(ei, NEDGES, ET, ebuf, emax, den);
    edge_msg_kernel<1, GCLASS><<<cdiv((long long)ET * (GCLASS / 4), TB), TB, 0, stream>>>(ei, NEDGES, ET, hA, ebuf, den, agg);
    finalize_kernel<false><<<cdiv((long long)NNODES * GCLASS, TB), TB, 0, stream>>>(agg, b3, out, NNODES, GCLASS);
  }
}